// LLGAE_67551245631805
// MI455X (gfx1250) — compile-verified
//
#include <hip/hip_runtime.h>

typedef float v2f __attribute__((ext_vector_type(2)));
typedef float v8f __attribute__((ext_vector_type(8)));

#define B_    1024
#define G_    256
#define K_    256
#define E_    64
#define BT    64          // batch rows per workgroup
#define BETA  0.05f
#define NSTEP 6

// LDS layouts (floats), padded strides chosen for conflict-free WMMA frag access
#define WT_S  272         // WT[e][g]  : e in [0,64),  g in [0,256)   (272 % 64 == 16)
#define WL_S  80          // Wl[g][e]  : g in [0,256), e in [0,64)    (80  % 64 == 16)
#define ZT_S  65          // ZT[e][b]  : e in [0,64),  b in [0,64)
#define DXT_S 65          // dxT[g][b] : g in [0,256), b in [0,64)

#define LDS_WT   0
#define LDS_WL   (LDS_WT  + 64  * WT_S)
#define LDS_ZT   (LDS_WL  + 256 * WL_S)
#define LDS_DXT  (LDS_ZT  + 64  * ZT_S)
#define LDS_MUK  (LDS_DXT + 256 * DXT_S)
#define LDS_SSQ  (LDS_MUK + 256)
#define LDS_TOT  (LDS_SSQ + 64)          // floats

// ---------------------------------------------------------------------------
// Kernel 1: per (k, batch-tile) run the 6-step fixed point with f32 WMMA.
// Computes dx' = -(x - mu - z W^T) = z W^T + (mu - x)  via D = A*B + C,
// z <- (1-beta) z - beta (dx' W).  Writes z (B,K,E) and ssq (B,K) to scratch.
// ---------------------------------------------------------------------------
__global__ __launch_bounds__(256) void lgae_iter_kernel(
    const float* __restrict__ images,   // (B,G)
    const float* __restrict__ mu,       // (G,K)
    const float* __restrict__ w,        // (K,G,E)
    float* __restrict__ zout,           // (B,K,E)
    float* __restrict__ ssq_out)        // (B,K)
{
    extern __shared__ float smem[];
    float* WT  = smem + LDS_WT;
    float* Wl  = smem + LDS_WL;
    float* ZT  = smem + LDS_ZT;
    float* dxT = smem + LDS_DXT;
    float* muk = smem + LDS_MUK;
    float* ssq = smem + LDS_SSQ;

    const int k   = blockIdx.x;
    const int b0  = blockIdx.y * BT;
    const int tid = threadIdx.x;
    const int lane = tid & 31;
    const int wv   = tid >> 5;     // wave id 0..7
    const int lo   = lane & 15;    // n / m-low index inside frag
    const int hi   = lane >> 4;    // half-wave select

    // ---- stage W (both orientations), mu column k, zero ZT / ssq -----------
    const float* wk = w + (size_t)k * (G_ * E_);
    for (int idx = tid; idx < G_ * E_; idx += 256) {
        float v = wk[idx];
        int g = idx >> 6, e = idx & 63;
        Wl[g * WL_S + e] = v;
        WT[e * WT_S + g] = v;
    }
    muk[tid] = mu[(size_t)tid * K_ + k];           // muT[k][g] = mu[g][k]
    for (int idx = tid; idx < 64 * ZT_S; idx += 256) ZT[idx] = 0.f;
    if (tid < BT) ssq[tid] = 0.f;
    __syncthreads();

    // ---- per-wave tile assignment ------------------------------------------
    const int r  = wv >> 1;        // batch row-group: rows [16r, 16r+16)
    const int h2 = wv & 1;         // GEMM1 column half (g), GEMM2 e half

    // (mu - x) fragments in D-layout, held in registers across all steps
    v8f xmn[8];
#pragma unroll
    for (int j = 0; j < 8; ++j) {
        const int g = 128 * h2 + 16 * j + lo;
        const float m_ = muk[g];
        v8f t;
#pragma unroll
        for (int v = 0; v < 8; ++v) {
            const int row = b0 + 16 * r + v + 8 * hi;
            t[v] = m_ - images[(size_t)row * G_ + g];
        }
        xmn[j] = t;
    }

    v8f zfrag[2];
#pragma unroll
    for (int j2 = 0; j2 < 2; ++j2) zfrag[j2] = v8f{};

    for (int it = 0; it < NSTEP; ++it) {
        // ---- GEMM1: dx' (64x256) = Z (64x64) * WT (64x256) + (mu - x) ------
        v8f acc[8];
#pragma unroll
        for (int j = 0; j < 8; ++j) acc[j] = xmn[j];

        for (int kk = 0; kk < E_; kk += 4) {
            v2f a;
            a.x = ZT[(kk + 2 * hi    ) * ZT_S + 16 * r + lo];
            a.y = ZT[(kk + 2 * hi + 1) * ZT_S + 16 * r + lo];
#pragma unroll
            for (int j = 0; j < 8; ++j) {
                const int g = 128 * h2 + 16 * j + lo;
                v2f bf;
                bf.x = WT[(kk + 2 * hi    ) * WT_S + g];
                bf.y = WT[(kk + 2 * hi + 1) * WT_S + g];
                acc[j] = __builtin_amdgcn_wmma_f32_16x16x4_f32(
                    false, a, false, bf, (short)0, acc[j], false, false);
            }
        }

        // scatter dx' into transposed LDS for GEMM2 A-frags
#pragma unroll
        for (int j = 0; j < 8; ++j) {
            const int g = 128 * h2 + 16 * j + lo;
#pragma unroll
            for (int v = 0; v < 8; ++v)
                dxT[g * DXT_S + 16 * r + v + 8 * hi] = acc[j][v];
        }

        if (it == NSTEP - 1) {      // squared-error partials (dx'^2 == dx^2)
#pragma unroll
            for (int v = 0; v < 8; ++v) {
                float p = 0.f;
#pragma unroll
                for (int j = 0; j < 8; ++j) p += acc[j][v] * acc[j][v];
                atomicAdd(&ssq[16 * r + v + 8 * hi], p);
            }
        }
        __syncthreads();

        // ---- GEMM2: t' (64x64) = dx' (64x256) * Wl (256x64) ----------------
        v8f tacc[2];
#pragma unroll
        for (int j2 = 0; j2 < 2; ++j2) tacc[j2] = v8f{};

        for (int kk = 0; kk < G_; kk += 4) {
            v2f a;
            a.x = dxT[(kk + 2 * hi    ) * DXT_S + 16 * r + lo];
            a.y = dxT[(kk + 2 * hi + 1) * DXT_S + 16 * r + lo];
#pragma unroll
            for (int j2 = 0; j2 < 2; ++j2) {
                const int e0 = 32 * h2 + 16 * j2 + lo;
                v2f bf;
                bf.x = Wl[(kk + 2 * hi    ) * WL_S + e0];
                bf.y = Wl[(kk + 2 * hi + 1) * WL_S + e0];
                tacc[j2] = __builtin_amdgcn_wmma_f32_16x16x4_f32(
                    false, a, false, bf, (short)0, tacc[j2], false, false);
            }
        }

        // ---- z update: z_new = (1-beta) z_old - beta * t' ------------------
        // each wave owns disjoint (e,b) positions of ZT -> no cross-wave hazard
#pragma unroll
        for (int j2 = 0; j2 < 2; ++j2) {
            const int ec = 32 * h2 + 16 * j2 + lo;
            v8f zn;
#pragma unroll
            for (int v = 0; v < 8; ++v) {
                const int brow = 16 * r + v + 8 * hi;
                const float zold = ZT[ec * ZT_S + brow];
                zn[v] = (1.f - BETA) * zold - BETA * tacc[j2][v];
            }
            zfrag[j2] = zn;
#pragma unroll
            for (int v = 0; v < 8; ++v)
                ZT[ec * ZT_S + 16 * r + v + 8 * hi] = zn[v];
        }
        __syncthreads();
    }

    // ---- outputs: z candidates and per-(b,k) squared error -----------------
#pragma unroll
    for (int j2 = 0; j2 < 2; ++j2) {
        const int e0 = 32 * h2 + 16 * j2;
#pragma unroll
        for (int v = 0; v < 8; ++v) {
            const int brow = b0 + 16 * r + v + 8 * hi;
            zout[((size_t)brow * K_ + k) * E_ + e0 + lo] = zfrag[j2][v];
        }
    }
    if (tid < BT) ssq_out[(size_t)(b0 + tid) * K_ + k] = ssq[tid];
}

// ---------------------------------------------------------------------------
// Kernel 2: per-b argmin over k of ssq (ties -> smallest k, matching argmax
// of the negated mean), then decode y[b][g] = sum_e z[b,kmax,e] w[kmax,g,e]
// + mu[g][kmax].
// ---------------------------------------------------------------------------
__global__ __launch_bounds__(256) void lgae_select_decode(
    const float* __restrict__ mu,       // (G,K)
    const float* __restrict__ w,        // (K,G,E)
    const float* __restrict__ zws,      // (B,K,E)
    const float* __restrict__ ssq,      // (B,K)
    float* __restrict__ y)              // (B,G)
{
    __shared__ unsigned long long best;
    __shared__ float zb[E_];
    __shared__ int ksel;

    const int b   = blockIdx.x;
    const int tid = threadIdx.x;

    if (tid == 0) best = ~0ULL;
    __syncthreads();

    {   // k = tid (blockDim == K_); pack value bits above index (val >= 0)
        const float v = ssq[(size_t)b * K_ + tid];
        const unsigned long long key =
            ((unsigned long long)__float_as_uint(v) << 32) | (unsigned int)tid;
        atomicMin(&best, key);
    }
    __syncthreads();
    if (tid == 0) ksel = (int)(best & 0xFFFFFFFFull);
    __syncthreads();
    const int kmax = ksel;

    if (tid < E_) zb[tid] = zws[((size_t)b * K_ + kmax) * E_ + tid];
    __syncthreads();

    const float4* w4 = (const float4*)(w + ((size_t)kmax * G_ + tid) * E_);
    const float4* z4 = (const float4*)zb;
    float acc = 0.f;
#pragma unroll
    for (int e4 = 0; e4 < E_ / 4; ++e4) {
        const float4 a = w4[e4];
        const float4 z = z4[e4];
        acc += a.x * z.x + a.y * z.y + a.z * z.z + a.w * z.w;
    }
    y[(size_t)b * G_ + tid] = acc + mu[(size_t)tid * K_ + kmax];
}

// ---------------------------------------------------------------------------
extern "C" void kernel_launch(void* const* d_in, const int* in_sizes, int n_in,
                              void* d_out, int out_size, void* d_ws, size_t ws_size,
                              hipStream_t stream) {
    (void)in_sizes; (void)n_in; (void)out_size; (void)ws_size;
    const float* images = (const float*)d_in[0];   // (B,G)
    const float* mu     = (const float*)d_in[1];   // (G,K)
    const float* w      = (const float*)d_in[2];   // (K,G,E)
    float* y = (float*)d_out;

    float* zws = (float*)d_ws;                        // B*K*E floats (64 MB)
    float* ssq = zws + (size_t)B_ * K_ * E_;          // B*K  floats (1 MB)

    const size_t shmem = (size_t)LDS_TOT * sizeof(float);   // ~236 KB < 320 KB/WGP
    dim3 grid1(K_, B_ / BT);
    lgae_iter_kernel<<<grid1, 256, shmem, stream>>>(images, mu, w, zws, ssq);
    lgae_select_decode<<<B_, 256, 0, stream>>>(mu, w, zws, ssq, y);
}